// TransformerBlock_44384192037436
// MI455X (gfx1250) — compile-verified
//
#include <hip/hip_runtime.h>
#include <math.h>

// Problem constants (match reference)
#define Bc     2
#define Nn     8192
#define Gg     64
#define DIMc   256
#define Hc     8
#define DHc    32
#define MLPc   1024
#define Mrows  (Bc * Nn)      // 16384 flattened rows
#define QKVW   768            // 3 * INNER

typedef __attribute__((ext_vector_type(2))) float v2f;
typedef __attribute__((ext_vector_type(8))) float v8f;

__device__ __forceinline__ float wave_sum32(float s) {
#pragma unroll
  for (int off = 16; off > 0; off >>= 1) s += __shfl_xor(s, off, 32);
  return s;
}

// ---------------------------------------------------------------------------
// Segment setup: starts[g] = lower_bound(batch, g); counts[g] = max(len, 1)
// ---------------------------------------------------------------------------
__global__ void seg_setup_kernel(const int* __restrict__ batch,
                                 int* __restrict__ starts,
                                 float* __restrict__ counts) {
  int t = threadIdx.x;
  if (t <= Gg) {
    int lo = 0, hi = Nn;
    while (lo < hi) {
      int mid = (lo + hi) >> 1;
      if (batch[mid] < t) lo = mid + 1; else hi = mid;
    }
    starts[t] = lo;
  }
  __syncthreads();
  if (t < Gg) counts[t] = fmaxf((float)(starts[t + 1] - starts[t]), 1.0f);
}

// ---------------------------------------------------------------------------
// fp32 WMMA GEMM: C[M,N] = A[M,K] @ W[K,N] (+ epilogue per MODE)
//   MODE 0: plain store                           (qkv)
//   MODE 1: + bias[col] + addend[r,c]             (z + x -> zadd)
//   MODE 2: A' = LN(A) inline; store gelu(+bias)  (ff)
//   MODE 3: + bias[col] + addend[r,c]             (final out)
// Block = 256 threads = 8 waves arranged 2x4; each wave owns a 32x64 block
// (2 row-tiles x 4 col-tiles -> 8 v8f accumulators). Per K-step:
// 2 b64 A loads + 8 b32 B loads feed 8 v_wmma_f32_16x16x4_f32.
// Grid: (N/256, M/64). All dims tile-aligned -> EXEC all-ones for WMMA.
// ---------------------------------------------------------------------------
template <int MODE>
__global__ __launch_bounds__(256)
void gemm_f32_wmma(const float* __restrict__ A, const float* __restrict__ W,
                   float* __restrict__ C, int M, int N, int K,
                   const float* __restrict__ bias,
                   const float* __restrict__ addend,
                   const float* __restrict__ mean,
                   const float* __restrict__ rstd,
                   const float* __restrict__ lnw,
                   const float* __restrict__ lnb) {
  const int lane    = threadIdx.x & 31;
  const int wave    = threadIdx.x >> 5;
  const int waverow = wave >> 2;             // 0..1
  const int wavecol = wave & 3;              // 0..3
  const int row0    = (blockIdx.y << 6) + (waverow << 5);   // 32 rows per wave
  const int col0    = (blockIdx.x << 8) + (wavecol << 6);   // 64 cols per wave
  const int khalf   = (lane >> 4) << 1;      // 0 for lanes 0-15, 2 for 16-31
  const int colL    = col0 + (lane & 15);
  const int arow0   = row0 + (lane & 15);
  const int arow1   = arow0 + 16;

  float m0 = 0.f, r0 = 1.f, m1 = 0.f, r1 = 1.f;
  if (MODE == 2) {
    m0 = mean[arow0]; r0 = rstd[arow0];
    m1 = mean[arow1]; r1 = rstd[arow1];
  }

  v8f acc[2][4];
#pragma unroll
  for (int rt = 0; rt < 2; ++rt)
#pragma unroll
    for (int ct = 0; ct < 4; ++ct) acc[rt][ct] = (v8f){};

  const float* Arow0 = A + (size_t)arow0 * K;
  const float* Arow1 = A + (size_t)arow1 * K;

  for (int k0 = 0; k0 < K; k0 += 4) {
    const int kk = k0 + khalf;
    // A fragments (16x4 f32): v2f per lane, contiguous 64-bit loads
    v2f a0; a0.x = Arow0[kk]; a0.y = Arow0[kk + 1];
    v2f a1; a1.x = Arow1[kk]; a1.y = Arow1[kk + 1];
    if (MODE == 2) {
      const float w0 = lnw[kk], w1 = lnw[kk + 1];
      const float c0 = lnb[kk], c1 = lnb[kk + 1];
      a0.x = (a0.x - m0) * r0 * w0 + c0;
      a0.y = (a0.y - m0) * r0 * w1 + c1;
      a1.x = (a1.x - m1) * r1 * w0 + c0;
      a1.y = (a1.y - m1) * r1 * w1 + c1;
    }
    const float* W0 = W + (size_t)kk * N;
#pragma unroll
    for (int ct = 0; ct < 4; ++ct) {
      const int c = colL + ct * 16;
      // B fragment (4x16 f32): VGPR0 = K{k0|k0+2}, VGPR1 = K{k0+1|k0+3}
      v2f b; b.x = W0[c]; b.y = W0[N + c];
      acc[0][ct] = __builtin_amdgcn_wmma_f32_16x16x4_f32(
          false, a0, false, b, (short)0, acc[0][ct], false, false);
      acc[1][ct] = __builtin_amdgcn_wmma_f32_16x16x4_f32(
          false, a1, false, b, (short)0, acc[1][ct], false, false);
    }
  }

  // C/D layout: VGPR i -> M = i (lanes 0-15) / M = 8+i (lanes 16-31)
  const int rsub = (lane >> 4) << 3;
#pragma unroll
  for (int rt = 0; rt < 2; ++rt) {
#pragma unroll
    for (int ct = 0; ct < 4; ++ct) {
      const int c = colL + ct * 16;
#pragma unroll
      for (int i = 0; i < 8; ++i) {
        const int r = row0 + rt * 16 + rsub + i;
        float v = acc[rt][ct][i];
        if (MODE == 1 || MODE == 3) {
          v += bias[c] + addend[(size_t)r * N + c];
        } else if (MODE == 2) {
          v += bias[c];
          v = 0.5f * v * (1.0f + erff(v * 0.70710678118654752440f));  // exact GELU
        }
        C[(size_t)r * N + c] = v;
      }
    }
  }
}

// ---------------------------------------------------------------------------
// Per-(row, head) 32-wide LayerNorm of K and V slices of qkv, in place.
// One wave32 per head-vector.
// ---------------------------------------------------------------------------
__global__ __launch_bounds__(256)
void ln_kv_kernel(float* __restrict__ qkv,
                  const float* __restrict__ ln1w, const float* __restrict__ ln1b,
                  const float* __restrict__ ln2w, const float* __restrict__ ln2b) {
  const int wid  = blockIdx.x * 8 + (threadIdx.x >> 5);
  const int lane = threadIdx.x & 31;
  const int row  = wid >> 3;
  const int h    = wid & 7;
  if (row >= Mrows) return;
  float* kp = qkv + (size_t)row * QKVW + DIMc + h * DHc;       // K slice
  float* vp = kp + DIMc;                                        // V slice
  {
    float x = kp[lane];
    float m = wave_sum32(x) * (1.0f / 32.0f);
    float d = x - m;
    float var = wave_sum32(d * d) * (1.0f / 32.0f);
    kp[lane] = d * rsqrtf(var + 1e-6f) * ln1w[lane] + ln1b[lane];
  }
  {
    float x = vp[lane];
    float m = wave_sum32(x) * (1.0f / 32.0f);
    float d = x - m;
    float var = wave_sum32(d * d) * (1.0f / 32.0f);
    vp[lane] = d * rsqrtf(var + 1e-6f) * ln2w[lane] + ln2b[lane];
  }
}

// ---------------------------------------------------------------------------
// ktv[g,b,h] = K_g^T V_g (32x32). One workgroup per (g,h,b); LDS-staged
// outer-product accumulation, 4 outputs per thread, no atomics.
// ---------------------------------------------------------------------------
__global__ __launch_bounds__(256)
void ktv_kernel(const float* __restrict__ qkv, const int* __restrict__ starts,
                float* __restrict__ ktv) {
  __shared__ float ks[8][32];
  __shared__ float vs[8][32];
  const int g = blockIdx.x, h = blockIdx.y, b = blockIdx.z;
  const int t  = threadIdx.x;
  const int d  = t >> 3;             // 0..31
  const int e4 = (t & 7) << 2;       // 0,4,...,28
  float a0 = 0.f, a1 = 0.f, a2 = 0.f, a3 = 0.f;
  const int s = starts[g], e = starts[g + 1];
  for (int base = s; base < e; base += 8) {
    const int cnt = min(8, e - base);
    __syncthreads();
    const int nd = t >> 5, ln = t & 31;
    if (nd < cnt) {
      const size_t ro = ((size_t)(b * Nn + base + nd)) * QKVW + h * DHc;
      ks[nd][ln] = qkv[ro + DIMc + ln];
      vs[nd][ln] = qkv[ro + 2 * DIMc + ln];
    }
    __syncthreads();
    for (int j = 0; j < cnt; ++j) {
      const float kd = ks[j][d];
      a0 = fmaf(kd, vs[j][e4 + 0], a0);
      a1 = fmaf(kd, vs[j][e4 + 1], a1);
      a2 = fmaf(kd, vs[j][e4 + 2], a2);
      a3 = fmaf(kd, vs[j][e4 + 3], a3);
    }
  }
  const size_t o = ((((size_t)g * Bc + b) * Hc + h) << 10) + d * 32 + e4;
  ktv[o + 0] = a0; ktv[o + 1] = a1; ktv[o + 2] = a2; ktv[o + 3] = a3;
}

// ---------------------------------------------------------------------------
// attn[row, h*32+e] = (1/count_g) * sum_d q[row,h,d] * ktv[g,b,h,d,e]
// One wave per (row, head): broadcast-q matvec, coalesced ktv row reads.
// ---------------------------------------------------------------------------
__global__ __launch_bounds__(256)
void attn_kernel(const float* __restrict__ qkv, const float* __restrict__ ktv,
                 const float* __restrict__ counts, const int* __restrict__ batch,
                 float* __restrict__ attn) {
  const int wid  = blockIdx.x * 8 + (threadIdx.x >> 5);
  const int lane = threadIdx.x & 31;
  const int h    = wid & 7;
  const int row  = wid >> 3;
  if (row >= Mrows) return;
  const int n = row & (Nn - 1);
  const int b = row >> 13;                 // row / 8192
  const int g = batch[n];
  const float inv = 1.0f / counts[g];
  const float q = qkv[(size_t)row * QKVW + h * DHc + lane] * inv;
  const float* kt = ktv + ((((size_t)g * Bc + b) * Hc + h) << 10);
  float acc = 0.f;
#pragma unroll
  for (int dd = 0; dd < 32; ++dd)
    acc = fmaf(__shfl(q, dd, 32), kt[dd * 32 + lane], acc);
  attn[(size_t)row * DIMc + h * DHc + lane] = acc;
}

// ---------------------------------------------------------------------------
// Per-row mean / rstd over DIM=256 of zadd (for inline LN in the MLP GEMM).
// ---------------------------------------------------------------------------
__global__ __launch_bounds__(256)
void lnstats_kernel(const float* __restrict__ zadd, float* __restrict__ mean,
                    float* __restrict__ rstd) {
  const int row  = blockIdx.x * 8 + (threadIdx.x >> 5);
  const int lane = threadIdx.x & 31;
  if (row >= Mrows) return;
  const float* p = zadd + (size_t)row * DIMc;
  float s = 0.f;
#pragma unroll
  for (int i = lane; i < DIMc; i += 32) s += p[i];
  const float m = wave_sum32(s) * (1.0f / DIMc);
  float s2 = 0.f;
#pragma unroll
  for (int i = lane; i < DIMc; i += 32) { float d = p[i] - m; s2 += d * d; }
  const float var = wave_sum32(s2) * (1.0f / DIMc);
  mean[row] = m;
  rstd[row] = rsqrtf(var + 1e-6f);
}

// ---------------------------------------------------------------------------
// Launch. Workspace layout (floats):
//   [0, 12582912)            qkv            (later reused as ff, part 1)
//   [12582912, 16777216)     attn           (later reused as ff, part 2)
//   [16777216, 20971520)     zadd
//   [20971520, 20987904)     mean
//   [20987904, 21004288)     rstd
//   [21004288, 22052864)     ktv
//   [22052864, 22052928)     counts
//   [22052928, 22052993)     starts (int)
// total ~88.2 MB
// ---------------------------------------------------------------------------
extern "C" void kernel_launch(void* const* d_in, const int* in_sizes, int n_in,
                              void* d_out, int out_size, void* d_ws, size_t ws_size,
                              hipStream_t stream) {
  const float* x     = (const float*)d_in[0];
  const int*   batch = (const int*)  d_in[1];
  const float* w_qkv = (const float*)d_in[2];
  const float* ln1w  = (const float*)d_in[3];
  const float* ln1b  = (const float*)d_in[4];
  const float* ln2w  = (const float*)d_in[5];
  const float* ln2b  = (const float*)d_in[6];
  const float* w_out = (const float*)d_in[7];
  const float* b_out = (const float*)d_in[8];
  const float* lnmw  = (const float*)d_in[9];
  const float* lnmb  = (const float*)d_in[10];
  const float* w1    = (const float*)d_in[11];
  const float* b1    = (const float*)d_in[12];
  const float* w2    = (const float*)d_in[13];
  const float* b2    = (const float*)d_in[14];
  float* out = (float*)d_out;

  float* ws     = (float*)d_ws;
  float* qkv    = ws;
  float* attn   = ws + 12582912;
  float* ff     = ws;                 // aliases qkv+attn (both dead by then)
  float* zadd   = ws + 16777216;
  float* meanb  = ws + 20971520;
  float* rstdb  = ws + 20987904;
  float* ktv    = ws + 21004288;
  float* counts = ws + 22052864;
  int*   starts = (int*)(ws + 22052928);

  seg_setup_kernel<<<1, 128, 0, stream>>>(batch, starts, counts);

  // qkv = x @ w_qkv                           [16384,256]x[256,768]
  gemm_f32_wmma<0><<<dim3(QKVW / 256, Mrows / 64), 256, 0, stream>>>(
      x, w_qkv, qkv, Mrows, QKVW, DIMc,
      nullptr, nullptr, nullptr, nullptr, nullptr, nullptr);

  ln_kv_kernel<<<Mrows * Hc / 8, 256, 0, stream>>>(qkv, ln1w, ln1b, ln2w, ln2b);

  ktv_kernel<<<dim3(Gg, Hc, Bc), 256, 0, stream>>>(qkv, starts, ktv);

  attn_kernel<<<Mrows * Hc / 8, 256, 0, stream>>>(qkv, ktv, counts, batch, attn);

  // zadd = attn @ w_out + b_out + x           [16384,256]x[256,256]
  gemm_f32_wmma<1><<<dim3(DIMc / 256, Mrows / 64), 256, 0, stream>>>(
      attn, w_out, zadd, Mrows, DIMc, DIMc,
      b_out, x, nullptr, nullptr, nullptr, nullptr);

  lnstats_kernel<<<Mrows / 8, 256, 0, stream>>>(zadd, meanb, rstdb);

  // ff = gelu(LN(zadd) @ w1 + b1)             [16384,256]x[256,1024]
  gemm_f32_wmma<2><<<dim3(MLPc / 256, Mrows / 64), 256, 0, stream>>>(
      zadd, w1, ff, Mrows, MLPc, DIMc,
      b1, nullptr, meanb, rstdb, lnmw, lnmb);

  // out = ff @ w2 + b2 + zadd                 [16384,1024]x[1024,256]
  gemm_f32_wmma<3><<<dim3(DIMc / 256, Mrows / 64), 256, 0, stream>>>(
      ff, w2, out, Mrows, DIMc, MLPc,
      b2, zadd, nullptr, nullptr, nullptr, nullptr);
}